// lstm_encoder_51625506898450
// MI455X (gfx1250) — compile-verified
//
#include <hip/hip_runtime.h>

typedef __attribute__((ext_vector_type(2))) float v2f;
typedef __attribute__((ext_vector_type(8))) float v8f;

#define BB 4096
#define LL 512
#define MM 17
#define HH 17
#define GG 68      // 4*H
#define KCH 9      // K chunks of 4 covering K = M+H = 34 (padded to 36)
#define NT 5       // N tiles of 16 covering 4H = 68 (padded to 80)

// One wave (32 lanes) per 16-row batch tile; runs the full 512-step recurrence
// with h/c resident in LDS, using f32 WMMA 16x16x4 for [x_t | h] @ [W_ih;W_hh]^T.
__global__ __launch_bounds__(32)
void lstm_fused_wmma(const float* __restrict__ X,
                     const float* __restrict__ Wih,
                     const float* __restrict__ Whh,
                     const float* __restrict__ bih,
                     const float* __restrict__ bhh,
                     float* __restrict__ out)
{
    __shared__ __align__(16) float AH[16][36];  // [x_t(17) | h(17) | pad(2)] per row
    __shared__ __align__(16) float GT[16][80];  // gates (padded to 80 cols)
    __shared__ float CB[16][17];                // cell state

    const int lane = threadIdx.x;
    const int hi   = lane >> 4;    // 0 or 1 (lane half)
    const int col  = lane & 15;    // N index within tile / A row
    const int row  = col;
    const int b0   = blockIdx.x * 16;

    // ---- Build resident B-matrix fragments (combined [W_ih; W_hh]^T, K=36 x N=80)
    // B 4x16 layout (mirror of documented A layout): lane half selects K pair.
    v2f  bf[NT][KCH];
    float bias[NT];
#pragma unroll
    for (int j = 0; j < NT; ++j) {
        const int n = 16 * j + col;
        bias[j] = (n < GG) ? (bih[n] + bhh[n]) : 0.0f;
#pragma unroll
        for (int kc = 0; kc < KCH; ++kc) {
            float w0 = 0.0f, w1 = 0.0f;
            if (n < GG) {
                const int k0 = 4 * kc + 2 * hi;
                if (k0 < MM)            w0 = Wih[n * MM + k0];
                else if (k0 < MM + HH)  w0 = Whh[n * HH + (k0 - MM)];
                const int k1 = k0 + 1;
                if (k1 < MM)            w1 = Wih[n * MM + k1];
                else if (k1 < MM + HH)  w1 = Whh[n * HH + (k1 - MM)];
            }
            v2f b; b.x = w0; b.y = w1;
            bf[j][kc] = b;
        }
    }

    // ---- init h = 0 (AH cols 17..35, includes zero K-padding) and c = 0
    for (int i = lane; i < 16 * 19; i += 32)
        AH[i / 19][MM + (i % 19)] = 0.0f;
    for (int i = lane; i < 16 * HH; i += 32)
        CB[i / HH][i % HH] = 0.0f;

    const float* xb = X + (size_t)b0 * LL * MM;

    for (int t = 0; t < LL; ++t) {
        // stage x tile for this timestep into the A panel
        for (int i = lane; i < 16 * MM; i += 32) {
            const int r = i / MM, m = i % MM;
            AH[r][m] = xb[(size_t)r * (LL * MM) + (size_t)t * MM + m];
        }
        __syncthreads();

        // A fragments from LDS (8B-aligned pair loads)
        v2f a[KCH];
#pragma unroll
        for (int kc = 0; kc < KCH; ++kc)
            a[kc] = *(const v2f*)&AH[row][4 * kc + 2 * hi];

        // gates = bias + [x|h] @ Wc   via chained f32 WMMA 16x16x4
        v8f acc[NT];
#pragma unroll
        for (int j = 0; j < NT; ++j) {
            v8f c0;
#pragma unroll
            for (int v = 0; v < 8; ++v) c0[v] = bias[j];
#pragma unroll
            for (int kc = 0; kc < KCH; ++kc)
                c0 = __builtin_amdgcn_wmma_f32_16x16x4_f32(
                        false, a[kc], false, bf[j][kc], (short)0, c0, false, false);
            acc[j] = c0;
        }

        // prefetch next timestep's x rows while math drains
        if (t + 1 < LL) {
            for (int i = lane; i < 16 * MM; i += 32) {
                const int r = i / MM, m = i % MM;
                __builtin_prefetch(&xb[(size_t)r * (LL * MM) + (size_t)(t + 1) * MM + m], 0, 3);
            }
        }

        // spill D fragments to LDS: D element (M = v + 8*hi, N = col)
#pragma unroll
        for (int j = 0; j < NT; ++j)
#pragma unroll
            for (int v = 0; v < 8; ++v)
                GT[v + 8 * hi][16 * j + col] = acc[j][v];
        __syncthreads();

        // gate nonlinearities + state update (torch gate order i,f,g,o)
        for (int i = lane; i < 16 * HH; i += 32) {
            const int r = i / HH, n = i % HH;
            const float ig = GT[r][n];
            const float fg = GT[r][HH + n];
            const float gg = GT[r][2 * HH + n];
            const float og = GT[r][3 * HH + n];
            const float si = 1.0f / (1.0f + __expf(-ig));
            const float sf = 1.0f / (1.0f + __expf(-fg));
            const float tg = 1.0f - 2.0f / (__expf(2.0f * gg) + 1.0f);
            const float so = 1.0f / (1.0f + __expf(-og));
            const float cn = sf * CB[r][n] + si * tg;
            const float hn = so * (1.0f - 2.0f / (__expf(2.0f * cn) + 1.0f));
            CB[r][n] = cn;
            AH[r][MM + n] = hn;  // feed next step's A panel
            out[(size_t)(b0 + r) * (LL * HH) + (size_t)t * HH + n] = hn;
        }
        __syncthreads();
    }

    // final (h, c) -> output sections 1 and 2
    const size_t offH = (size_t)BB * LL * HH;
    const size_t offC = offH + (size_t)BB * HH;
    for (int i = lane; i < 16 * HH; i += 32) {
        const int r = i / HH, n = i % HH;
        const size_t g = (size_t)(b0 + r) * HH + n;
        out[offH + g] = AH[r][MM + n];
        out[offC + g] = CB[r][n];
    }
}

// Stream the x echo into the output tail (contiguous, float4, grid-stride).
__global__ void copy_x_kernel(const float4* __restrict__ src,
                              float4* __restrict__ dst, int n4)
{
    int i = blockIdx.x * blockDim.x + threadIdx.x;
    const int stride = gridDim.x * blockDim.x;
    for (; i < n4; i += stride)
        dst[i] = src[i];
}

extern "C" void kernel_launch(void* const* d_in, const int* in_sizes, int n_in,
                              void* d_out, int out_size, void* d_ws, size_t ws_size,
                              hipStream_t stream)
{
    (void)in_sizes; (void)n_in; (void)out_size; (void)d_ws; (void)ws_size;
    const float* X   = (const float*)d_in[0];
    const float* Wih = (const float*)d_in[1];
    const float* Whh = (const float*)d_in[2];
    const float* bih = (const float*)d_in[3];
    const float* bhh = (const float*)d_in[4];
    float* out = (float*)d_out;

    lstm_fused_wmma<<<BB / 16, 32, 0, stream>>>(X, Wih, Whh, bih, bhh, out);

    const size_t offX = (size_t)BB * LL * HH + 2ull * BB * HH;
    const int n4 = (BB * LL * MM) / 4;   // 35,651,584 / 4
    copy_x_kernel<<<4096, 256, 0, stream>>>((const float4*)X,
                                            (float4*)(out + offX), n4);
}